// Transformer_56788057588499
// MI455X (gfx1250) — compile-verified
//
#include <hip/hip_runtime.h>
#include <cstdint>

#define VOCAB   32000
#define D_MODEL 512
#define BATCH   8
#define SEQ     4096

// angle(s,d) = s * 10000^(-2d/D) = s * exp2(-d * 2*log2(10000)/D)
// K = 2*log2(10000)/512 = 13.287712379549449 / 256
#define PE_K 0.05190512648261504f

// Native vector types (HIP's float4 is a class; builtins need real vectors)
typedef float f4  __attribute__((ext_vector_type(4)));
typedef int   v4i __attribute__((ext_vector_type(4)));

typedef __attribute__((address_space(1))) v4i* g_v4i_ptr;   // global
typedef __attribute__((address_space(3))) v4i* l_v4i_ptr;   // LDS

// ---- CDNA5 async global->LDS copy (b128) ----
__device__ __forceinline__ void async_copy_b128(const float* gsrc, float* lds_dst) {
#if __has_builtin(__builtin_amdgcn_global_load_async_to_lds_b128)
    __builtin_amdgcn_global_load_async_to_lds_b128(
        (g_v4i_ptr)(v4i*)(float*)gsrc, (l_v4i_ptr)(v4i*)lds_dst, 0, 0);
#else
    unsigned lofs = (unsigned)(size_t)((__attribute__((address_space(3))) char*)lds_dst);
    asm volatile("global_load_async_to_lds_b128 %0, %1, off"
                 :: "v"(lofs), "v"(gsrc) : "memory");
#endif
}

__device__ __forceinline__ void wait_asynccnt0() {
#if __has_builtin(__builtin_amdgcn_s_wait_asynccnt)
    __builtin_amdgcn_s_wait_asynccnt(0);
#else
    asm volatile("s_wait_asynccnt 0" ::: "memory");
#endif
}

__device__ __forceinline__ float fast_exp2(float x) {
#if __has_builtin(__builtin_amdgcn_exp2f)
    return __builtin_amdgcn_exp2f(x);
#else
    return exp2f(x);
#endif
}

// One 128-thread block per sequence position s.
// All BATCH rows at this s share one PE row; row loop is wave-uniform so the
// async loads / stores use scalar-base + per-lane-offset addressing.
__global__ __launch_bounds__(128) void emb_pe_kernel(
    const int*   __restrict__ tokens,     // [BATCH, SEQ] int32
    const float* __restrict__ emb_table,  // [VOCAB, D_MODEL] f32
    float*       __restrict__ out)        // [BATCH, SEQ, D_MODEL] f32
{
    __shared__ float embLds[BATCH][D_MODEL];   // 16 KiB
    __shared__ float peLds[D_MODEL];           //  2 KiB

    const int s   = blockIdx.x;
    const int tid = threadIdx.x;
    const int col = tid << 2;             // float index 0..508, 16B aligned

    // Uniform (per-block) token ids -> scalar loads
    int toks[BATCH];
#pragma unroll
    for (int b = 0; b < BATCH; ++b)
        toks[b] = tokens[b * SEQ + s];

    // ---- Phase 1: fire async DMA of the 8 gathered embedding rows into LDS.
    // 128 lanes x b128 = one 512-float row per iteration; row base is uniform.
#pragma unroll
    for (int row = 0; row < BATCH; ++row) {
        const float* gsrc = emb_table + (size_t)toks[row] * D_MODEL + col;
        async_copy_b128(gsrc, &embLds[row][col]);
    }

    // ---- Phase 2: compute PE row for this s (overlaps the async DMA).
    // Reference semantics: exponent uses 2*i/D for EVERY i; even i -> sin, odd -> cos.
    {
        const float fs = (float)s;
        f4 pe;
        pe.x = sinf(fs * fast_exp2(-(float)(col + 0) * PE_K));   // even -> sin
        pe.y = cosf(fs * fast_exp2(-(float)(col + 1) * PE_K));   // odd  -> cos
        pe.z = sinf(fs * fast_exp2(-(float)(col + 2) * PE_K));
        pe.w = cosf(fs * fast_exp2(-(float)(col + 3) * PE_K));
        *(f4*)&peLds[col] = pe;
    }

    // ---- Phase 3: drain async DMA (per-wave) then barrier (cross-wave + PE stores).
    wait_asynccnt0();
    __syncthreads();

    // ---- Phase 4: add PE and stream out with non-temporal b128 stores
    // (keep the L2-resident embedding table hot; output is write-once).
    const f4 p = *(const f4*)&peLds[col];
#pragma unroll
    for (int row = 0; row < BATCH; ++row) {
        f4 v = *(const f4*)&embLds[row][col];
        v += p;
        f4* dst = (f4*)(out + ((size_t)row * SEQ + s) * D_MODEL + col);
        __builtin_nontemporal_store(v, dst);
    }
}

extern "C" void kernel_launch(void* const* d_in, const int* in_sizes, int n_in,
                              void* d_out, int out_size, void* d_ws, size_t ws_size,
                              hipStream_t stream) {
    const int*   tokens    = (const int*)d_in[0];    // [8, 4096] int32
    const float* emb_table = (const float*)d_in[1];  // [32000, 512] f32
    float*       out       = (float*)d_out;          // [8, 4096, 512] f32

    emb_pe_kernel<<<SEQ, 128, 0, stream>>>(tokens, emb_table, out);
}